// DenseRelationModule_50251117363741
// MI455X (gfx1250) — compile-verified
//
#include <hip/hip_runtime.h>
#include <math.h>

// Problem constants (from reference)
#define BB     2
#define NN     512
#define OBJ_D  320
#define LANG_D 256
#define GEO_D  6
#define HID    256

typedef __attribute__((ext_vector_type(2))) float v2f;
typedef __attribute__((ext_vector_type(8))) float v8f;

// ---- f32 WMMA 16x16x4 (CDNA5 native fp32 matrix op) -------------------------
// 8-arg pattern: (neg_a, A, neg_b, B, c_mod, C, reuse_a, reuse_b)
__device__ __forceinline__ v8f wmma_f32_k4(v2f a, v2f b, v8f c) {
    return __builtin_amdgcn_wmma_f32_16x16x4_f32(
        /*neg_a=*/false, a, /*neg_b=*/false, b,
        /*c_mod=*/(short)0, c, /*reuse_a=*/false, /*reuse_b=*/false);
}

// ---- CDNA5 async global->LDS copy (no VGPR round trip, ASYNCcnt-tracked) ----
// vdst = VGPR holding LDS byte address (low 32 bits of a generic shared ptr),
// vaddr = 64-bit global address. 16 bytes per active lane.
__device__ __forceinline__ void async_g2l_b128(void* lds_dst, const void* gsrc) {
    unsigned l = (unsigned)(unsigned long long)lds_dst;  // LDS byte offset
    asm volatile("global_load_async_to_lds_b128 %0, %1, off"
                 :: "v"(l), "v"(gsrc)
                 : "memory");
}
__device__ __forceinline__ void async_wait0() {
    asm volatile("s_wait_asynccnt 0x0" ::: "memory");
}

__device__ __forceinline__ float gelu_exact(float x) {
    // exact gelu: 0.5*x*(1+erf(x/sqrt(2)))
    return 0.5f * x * (1.0f + erff(x * 0.70710678118654752f));
}

// ============================================================================
// Kernel 0: row-wise L2 normalize object embeddings. One block per (b,n) row.
// ============================================================================
__global__ __launch_bounds__(128)
void k_norm(const float* __restrict__ emb, float* __restrict__ emb_n) {
    const int row = blockIdx.x;                    // 0 .. B*N-1
    const float* src = emb + (size_t)row * OBJ_D;
    float* dst = emb_n + (size_t)row * OBJ_D;
    __shared__ float red[128];
    float s = 0.0f;
    for (int k = threadIdx.x; k < OBJ_D; k += 128) { float v = src[k]; s += v * v; }
    red[threadIdx.x] = s;
    __syncthreads();
    for (int off = 64; off > 0; off >>= 1) {
        if (threadIdx.x < off) red[threadIdx.x] += red[threadIdx.x + off];
        __syncthreads();
    }
    const float inv = 1.0f / fmaxf(sqrtf(red[0]), 1e-12f);
    for (int k = threadIdx.x; k < OBJ_D; k += 128) dst[k] = src[k] * inv;
}

// ============================================================================
// Kernel 1: hl[b,h] = l2norm(lang)[b] @ Wl + b1.   One block per batch.
// ============================================================================
__global__ __launch_bounds__(256)
void k_hl(const float* __restrict__ lang, const float* __restrict__ W1,
          const float* __restrict__ b1, float* __restrict__ hl) {
    const int b = blockIdx.x;
    const int h = threadIdx.x;               // 0..255 == LANG_D == HID
    __shared__ float sl[LANG_D];
    __shared__ float red[256];
    const float v = lang[b * LANG_D + h];
    sl[h] = v;
    red[h] = v * v;
    __syncthreads();
    for (int off = 128; off > 0; off >>= 1) {
        if (h < off) red[h] += red[h + off];
        __syncthreads();
    }
    const float inv = 1.0f / fmaxf(sqrtf(red[0]), 1e-12f);
    const float* Wl = W1 + (size_t)(2 * OBJ_D + GEO_D) * HID;   // rows 646..901
    float acc = b1[h];
    #pragma unroll 4
    for (int l = 0; l < LANG_D; ++l)
        acc = fmaf(sl[l] * inv, Wl[l * HID + h], acc);
    hl[b * HID + h] = acc;
}

// ============================================================================
// Kernel 2: hi/hj GEMMs via f32 WMMA 16x16x4.
// grid.x = (B*N)/16 row tiles; block = 512 (16 waves); wave w owns 16-col tile.
// A tile (16 x 320, normalized emb) staged in padded LDS via async B128 copies.
// hi = emb_n@Wi - geom@Wg ; hj = emb_n@Wj + geom@Wg (geom part via VALU, K=6).
// ============================================================================
__global__ __launch_bounds__(512)
void k_hij(const float* __restrict__ emb_n, const float* __restrict__ geom,
           const float* __restrict__ W1, float* __restrict__ hi,
           float* __restrict__ hj) {
    constexpr int AS = OBJ_D + 4;            // pad -> conflict-free LDS reads
    __shared__ __align__(16) float sA[16 * AS];   // ~20.7 KB
    const int row0 = blockIdx.x * 16;

    // 16 rows x 80 float4 chunks = 1280 chunks, direct global->LDS
    for (int idx = threadIdx.x; idx < 16 * (OBJ_D / 4); idx += 512) {
        const int r = idx / (OBJ_D / 4), c = idx - r * (OBJ_D / 4);
        async_g2l_b128(&sA[r * AS + c * 4],
                       &emb_n[(size_t)(row0 + r) * OBJ_D + c * 4]);
    }
    async_wait0();
    __syncthreads();

    const int wave = threadIdx.x >> 5;
    const int lane = threadIdx.x & 31;
    const int col0 = wave * 16;              // 16 waves x 16 = 256 = HID
    const int m    = lane & 15;              // A row / B column within tile
    const int kb   = (lane >> 4) * 2;        // K sub-pair per half-wave

    const float* Wi = W1;                                   // rows [0,320)
    const float* Wj = W1 + (size_t)OBJ_D * HID;             // rows [320,640)
    v8f ai = {}, aj = {};

    #pragma unroll 4
    for (int k0 = 0; k0 < OBJ_D; k0 += 4) {
        const int k = k0 + kb;
        v2f a = *(const v2f*)&sA[m * AS + k];               // 8B-aligned LDS read
        v2f bi, bj;
        bi.x = Wi[(k + 0) * HID + col0 + m];
        bi.y = Wi[(k + 1) * HID + col0 + m];
        bj.x = Wj[(k + 0) * HID + col0 + m];
        bj.y = Wj[(k + 1) * HID + col0 + m];
        ai = wmma_f32_k4(a, bi, ai);
        aj = wmma_f32_k4(a, bj, aj);
    }

    // geom correction (K = 6) + store, in C-matrix layout:
    // VGPR r -> M = r + 8*(lane>=16), N = col0 + (lane&15)
    const int n = col0 + m;
    const float* Wg = W1 + (size_t)(2 * OBJ_D) * HID;       // rows [640,646)
    #pragma unroll
    for (int r = 0; r < 8; ++r) {
        const int row = row0 + r + ((lane >> 4) << 3);
        float g = 0.0f;
        #pragma unroll
        for (int t = 0; t < GEO_D; ++t)
            g = fmaf(geom[row * GEO_D + t], Wg[t * HID + n], g);
        hi[(size_t)row * HID + n] = ai[r] - g;
        hj[(size_t)row * HID + n] = aj[r] + g;
    }
}

// ============================================================================
// Kernel 3: fused pair scoring. scores[b,i,j] = W2 . gelu(hi[i]+hj[j]+hl) + b2
// One block per 16x16 (i,j) tile; one thread per pair; hi/hj tiles staged into
// padded LDS via async B128 copies (conflict-free j-tile reads).
// ============================================================================
__global__ __launch_bounds__(256)
void k_scores(const float* __restrict__ hi, const float* __restrict__ hj,
              const float* __restrict__ hl, const float* __restrict__ W2,
              const float* __restrict__ b2, float* __restrict__ scores) {
    constexpr int S = HID + 4;
    __shared__ __align__(16) float shi[16 * S];
    __shared__ __align__(16) float shj[16 * S];
    __shared__ float shl[HID];
    __shared__ float sw2[HID];

    const int b  = blockIdx.z;
    const int i0 = blockIdx.y * 16;
    const int j0 = blockIdx.x * 16;
    const size_t base = (size_t)b * NN * HID;

    // 16 rows x 64 float4 chunks per tile, direct global->LDS
    for (int idx = threadIdx.x; idx < 16 * (HID / 4); idx += 256) {
        const int r = idx >> 6, c = idx & 63;
        async_g2l_b128(&shi[r * S + c * 4], &hi[base + (size_t)(i0 + r) * HID + c * 4]);
        async_g2l_b128(&shj[r * S + c * 4], &hj[base + (size_t)(j0 + r) * HID + c * 4]);
    }
    shl[threadIdx.x] = hl[b * HID + threadIdx.x];
    sw2[threadIdx.x] = W2[threadIdx.x];
    async_wait0();
    __syncthreads();

    const int il = threadIdx.x >> 4;
    const int jl = threadIdx.x & 15;
    const float* pi = &shi[il * S];
    const float* pj = &shj[jl * S];
    float acc = 0.0f;
    #pragma unroll 4
    for (int h = 0; h < HID; ++h) {
        const float x = pi[h] + pj[h] + shl[h];
        acc = fmaf(gelu_exact(x), sw2[h], acc);
    }
    scores[((size_t)(b * NN + i0 + il)) * NN + (j0 + jl)] = acc + b2[0];
}

// ============================================================================
// Kernel 4: row softmax over j (512) -> P (in place) + relation_scores.
// One block per (b,i) row; 256 threads, 2 elements each.
// ============================================================================
__global__ __launch_bounds__(256)
void k_softmax(float* __restrict__ sc, float* __restrict__ rs) {
    const int row = blockIdx.x;                 // b*N + i
    float* s = sc + (size_t)row * NN;
    const int t = threadIdx.x;
    __shared__ float red[256];

    const float a0 = s[t], a1 = s[t + 256];
    red[t] = fmaxf(a0, a1);
    __syncthreads();
    for (int off = 128; off > 0; off >>= 1) {
        if (t < off) red[t] = fmaxf(red[t], red[t + off]);
        __syncthreads();
    }
    const float mx = red[0];
    __syncthreads();

    const float e0 = expf(a0 - mx), e1 = expf(a1 - mx);
    red[t] = e0 + e1;
    __syncthreads();
    for (int off = 128; off > 0; off >>= 1) {
        if (t < off) red[t] += red[t + off];
        __syncthreads();
    }
    const float denom = red[0];
    __syncthreads();

    red[t] = e0 * a0 + e1 * a1;                 // sum p*s (unnormalized)
    __syncthreads();
    for (int off = 128; off > 0; off >>= 1) {
        if (t < off) red[t] += red[t + off];
        __syncthreads();
    }
    const float inv = 1.0f / denom;
    s[t]       = e0 * inv;
    s[t + 256] = e1 * inv;
    if (t == 0) rs[row] = red[0] * inv;
}

// ============================================================================
// Kernel 5: relation_context = P @ emb_n via f32 WMMA 16x16x4.
// grid = (N/16 row tiles, B); block = 640 (20 waves -> 20 x 16 = 320 = OBJ_D).
// P tile (16 x 512) staged in padded LDS via async B128 copies.
// ============================================================================
__global__ __launch_bounds__(640)
void k_ctx(const float* __restrict__ P, const float* __restrict__ emb_n,
           float* __restrict__ ctx) {
    constexpr int PS = NN + 4;
    __shared__ __align__(16) float sP[16 * PS];   // ~33 KB
    const int b  = blockIdx.y;
    const int i0 = blockIdx.x * 16;
    const float* Pb = P + (size_t)b * NN * NN;

    // 16 rows x 128 float4 chunks = 2048 chunks, direct global->LDS
    for (int idx = threadIdx.x; idx < 16 * (NN / 4); idx += 640) {
        const int r = idx >> 7, c = idx & 127;
        async_g2l_b128(&sP[r * PS + c * 4], &Pb[(size_t)(i0 + r) * NN + c * 4]);
    }
    async_wait0();
    __syncthreads();

    const int wave = threadIdx.x >> 5;          // 0..19
    const int lane = threadIdx.x & 31;
    const int d0   = wave * 16;
    const int m    = lane & 15;
    const int kb   = (lane >> 4) * 2;

    const float* Eb = emb_n + (size_t)b * NN * OBJ_D;
    v8f acc = {};
    #pragma unroll 4
    for (int k0 = 0; k0 < NN; k0 += 4) {
        const int k = k0 + kb;
        v2f a = *(const v2f*)&sP[m * PS + k];
        v2f bb;
        bb.x = Eb[(size_t)(k + 0) * OBJ_D + d0 + m];
        bb.y = Eb[(size_t)(k + 1) * OBJ_D + d0 + m];
        acc = wmma_f32_k4(a, bb, acc);
    }
    #pragma unroll
    for (int r = 0; r < 8; ++r) {
        const int row = i0 + r + ((lane >> 4) << 3);
        ctx[((size_t)(b * NN + row)) * OBJ_D + d0 + m] = acc[r];
    }
}

// ============================================================================
extern "C" void kernel_launch(void* const* d_in, const int* in_sizes, int n_in,
                              void* d_out, int out_size, void* d_ws, size_t ws_size,
                              hipStream_t stream) {
    const float* emb  = (const float*)d_in[0];   // (B,N,320)
    const float* geo  = (const float*)d_in[1];   // (B,N,6)
    const float* utt  = (const float*)d_in[2];   // (B,256)
    const float* W1   = (const float*)d_in[3];   // (902,256)
    const float* b1   = (const float*)d_in[4];   // (256,)
    const float* W2   = (const float*)d_in[5];   // (256,)
    const float* b2   = (const float*)d_in[6];   // scalar

    // Workspace carve-up (floats): 5.25 MB total
    float* ws     = (float*)d_ws;
    float* emb_n  = ws;                                  // B*N*OBJ_D = 327680
    float* hi     = emb_n + (size_t)BB * NN * OBJ_D;     // B*N*HID   = 262144
    float* hj     = hi    + (size_t)BB * NN * HID;       //            262144
    float* hl     = hj    + (size_t)BB * NN * HID;       // B*HID     = 512
    float* scores = hl    + (size_t)BB * HID;            // B*N*N     = 524288

    float* rs  = (float*)d_out;          // relation_scores (B,N) first
    float* ctx = rs + BB * NN;           // relation_context (B,N,320)

    k_norm<<<BB * NN, 128, 0, stream>>>(emb, emb_n);
    k_hl<<<BB, 256, 0, stream>>>(utt, W1, b1, hl);
    k_hij<<<(BB * NN) / 16, 512, 0, stream>>>(emb_n, geo, W1, hi, hj);
    dim3 g2(NN / 16, NN / 16, BB);
    k_scores<<<g2, 256, 0, stream>>>(hi, hj, hl, W2, b2, scores);
    k_softmax<<<BB * NN, 256, 0, stream>>>(scores, rs);
    dim3 g4(NN / 16, BB);
    k_ctx<<<g4, 640, 0, stream>>>(scores, emb_n, ctx);
}